// VariableSelectionNetwork_75505525064140
// MI455X (gfx1250) — compile-verified
//
#include <hip/hip_runtime.h>
#include <math.h>

// VariableSelectionNetwork fused kernel for MI455X (gfx1250, wave32).
// B=8, S=2048, F=32, D=256.
// out layout: [B*S*D] float32 then weights [B*S*F] float32.

typedef __attribute__((ext_vector_type(2))) float v2f;
typedef __attribute__((ext_vector_type(8))) float v8f;

#define BTOK 16          // tokens per block (WMMA M tile)
#define FDIM 32
#define DDIM 256
#define SELP 260         // padded pitch for selected tile in LDS
#define APITCH 33        // padded pitch for A fragments in LDS
#define NTOK (8 * 2048)  // B*S

// ---------------------------------------------------------------------------
// Kernel 1: per-feature closed-form LayerNorm stats of z = x*W[f,:] + b[f,:]
//   mu(x)  = x*mW + mB
//   var(x) = x^2*vW + 2x*cWB + vB   (always >= 0 in exact arithmetic)
// ---------------------------------------------------------------------------
__global__ __launch_bounds__(32) void vsn_stats(const float* __restrict__ W,
                                                const float* __restrict__ Bv,
                                                float* __restrict__ st) {
  int f = threadIdx.x;  // exactly 32 lanes, one feature each
  float sw = 0.f, sb = 0.f, sww = 0.f, swb = 0.f, sbb = 0.f;
  for (int d = 0; d < DDIM; ++d) {
    float w = W[f * DDIM + d];
    float b = Bv[f * DDIM + d];
    sw += w; sb += b; sww += w * w; swb += w * b; sbb += b * b;
  }
  const float inv = 1.0f / (float)DDIM;
  float mW = sw * inv, mB = sb * inv;
  st[f * 8 + 0] = mW;
  st[f * 8 + 1] = mB;
  st[f * 8 + 2] = sww * inv - mW * mW;   // var(W[f,:])
  st[f * 8 + 3] = swb * inv - mW * mB;   // cov(W,b)
  st[f * 8 + 4] = sbb * inv - mB * mB;   // var(b[f,:])
}

// ---------------------------------------------------------------------------
// Kernel 2: fused main kernel. One block handles 16 tokens.
//   Phase A: scores[t,f] = sum_d gelu(LN(z)[d]*g+beta) * score_w[d]  (wave/pair)
//   Phase B: softmax over F per token (one wave per token, shfl_xor, wave32)
//   Phase C: selected = (w*x) @ W + w @ b  via v_wmma_f32_16x16x4_f32
//   Phase D: final LayerNorm over D, write out
// ---------------------------------------------------------------------------
__global__ __launch_bounds__(256) void vsn_main(
    const float* __restrict__ x, const float* __restrict__ W,
    const float* __restrict__ Bv, const float* __restrict__ ln1g,
    const float* __restrict__ ln1b, const float* __restrict__ scw,
    const float* __restrict__ scb, const float* __restrict__ outg,
    const float* __restrict__ outb, const float* __restrict__ st,
    float* __restrict__ out, float* __restrict__ wout) {
  __shared__ float sX[BTOK][FDIM];      // x for this token tile
  __shared__ float sA1[BTOK][APITCH];   // weights * x   (A-matrix for W GEMM)
  __shared__ float sA2[BTOK][APITCH];   // weights       (A-matrix for b GEMM)
  __shared__ float sSC[BTOK][FDIM];     // scores
  __shared__ float sSel[BTOK][SELP];    // selected (WMMA D tiles)
  __shared__ float sG[DDIM], sBt[DDIM], sSw[DDIM], sOG[DDIM], sOB[DDIM];

  const int tid = threadIdx.x;
  const int lane = tid & 31;
  const int wave = tid >> 5;
  const long tok0 = (long)blockIdx.x * BTOK;

  // ---- stage small per-D vectors + token x slab ----
  for (int i = tid; i < DDIM; i += 256) {
    sG[i] = ln1g[i]; sBt[i] = ln1b[i]; sSw[i] = scw[i];
    sOG[i] = outg[i]; sOB[i] = outb[i];
  }
  for (int i = tid; i < BTOK * FDIM; i += 256)
    sX[i >> 5][i & 31] = x[tok0 * FDIM + i];
  __syncthreads();

  const float score_bias = scb[0];
  const float ISQRT2 = 0.70710678118654752f;

  // ---- Phase A: gelu/score reduction; one wave per (t,f) pair, lanes over d
  for (int p = wave; p < BTOK * FDIM; p += 8) {
    int t = p >> 5, f = p & 31;
    float xv = sX[t][f];
    float mW = st[f * 8 + 0], mB = st[f * 8 + 1];
    float vW = st[f * 8 + 2], cWB = st[f * 8 + 3], vB = st[f * 8 + 4];
    float mu = fmaf(xv, mW, mB);
    float var = fmaf(xv * xv, vW, fmaf(2.0f * xv, cWB, vB));
    float rstd = rsqrtf(var + 1e-5f);
    float acc = 0.0f;
#pragma unroll
    for (int j = 0; j < 8; ++j) {
      int d = lane + 32 * j;                        // coalesced, L2-resident
      float z = fmaf(xv, W[f * DDIM + d], Bv[f * DDIM + d]);
      float zh = (z - mu) * rstd;
      float y = fmaf(zh, sG[d], sBt[d]);
      float gl = 0.5f * y * (1.0f + erff(y * ISQRT2));  // exact gelu
      acc = fmaf(gl, sSw[d], acc);
    }
    for (int off = 16; off > 0; off >>= 1) acc += __shfl_xor(acc, off, 32);
    if (lane == 0) sSC[t][f] = acc + score_bias;
  }
  __syncthreads();

  // ---- Phase B: softmax over F (=32 = one wave32); lane == f ----
  for (int t = wave; t < BTOK; t += 8) {
    float v = sSC[t][lane];
    float m = v;
    for (int off = 16; off > 0; off >>= 1) m = fmaxf(m, __shfl_xor(m, off, 32));
    float e = expf(v - m);
    float s = e;
    for (int off = 16; off > 0; off >>= 1) s += __shfl_xor(s, off, 32);
    float wgt = e / s;
    sA2[t][lane] = wgt;
    sA1[t][lane] = wgt * sX[t][lane];
    wout[(tok0 + t) * FDIM + lane] = wgt;   // second output tensor
  }
  __syncthreads();

  // ---- Phase C: selected[16,256] = A1 @ W + A2 @ b via WMMA f32 16x16x4 ----
  // A 16x4 f32 layout (ISA 7.12.2): lanes 0-15 rows M=0..15, VGPR0 holds
  // K=k0 (lo half) / K=k0+2 (hi half); VGPR1 holds K=k0+1 / K=k0+3.
  // B 4x16 mirrors it: VGPR0 row K=k0 (lo) / K=k0+2 (hi), N striped on lanes.
  {
    const int row = lane & 15;
    const int hi = lane >> 4;
    for (int sub = 0; sub < 2; ++sub) {
      const int n0 = wave * 32 + sub * 16;   // 8 waves x 2 = 16 N-tiles = D
      v8f c = {0.f, 0.f, 0.f, 0.f, 0.f, 0.f, 0.f, 0.f};
#pragma unroll
      for (int kk = 0; kk < FDIM; kk += 4) {   // W-part, A = weights*x
        int k0 = kk + 2 * hi;
        v2f a, bf;
        a.x = sA1[row][k0];
        a.y = sA1[row][k0 + 1];
        bf.x = W[k0 * DDIM + n0 + row];
        bf.y = W[(k0 + 1) * DDIM + n0 + row];
        c = __builtin_amdgcn_wmma_f32_16x16x4_f32(false, a, false, bf,
                                                  (short)0, c, false, false);
      }
#pragma unroll
      for (int kk = 0; kk < FDIM; kk += 4) {   // b-part, A = weights
        int k0 = kk + 2 * hi;
        v2f a, bf;
        a.x = sA2[row][k0];
        a.y = sA2[row][k0 + 1];
        bf.x = Bv[k0 * DDIM + n0 + row];
        bf.y = Bv[(k0 + 1) * DDIM + n0 + row];
        c = __builtin_amdgcn_wmma_f32_16x16x4_f32(false, a, false, bf,
                                                  (short)0, c, false, false);
      }
      // D layout: VGPR r -> (M = r + 8*hi, N = row)
#pragma unroll
      for (int r = 0; r < 8; ++r) sSel[r + 8 * hi][n0 + row] = c[r];
    }
  }
  __syncthreads();

  // ---- Phase D: final LayerNorm over D per token, write out ----
  for (int t = wave; t < BTOK; t += 8) {
    float s1 = 0.f, s2 = 0.f;
#pragma unroll
    for (int j = 0; j < 8; ++j) {
      float v = sSel[t][lane + 32 * j];
      s1 += v; s2 += v * v;
    }
    for (int off = 16; off > 0; off >>= 1) {
      s1 += __shfl_xor(s1, off, 32);
      s2 += __shfl_xor(s2, off, 32);
    }
    float mu = s1 * (1.0f / (float)DDIM);
    float var = s2 * (1.0f / (float)DDIM) - mu * mu;
    float rstd = rsqrtf(var + 1e-5f);
    long base = (tok0 + t) * (long)DDIM;
#pragma unroll
    for (int j = 0; j < 8; ++j) {
      int d = lane + 32 * j;
      float v = sSel[t][d];
      out[base + d] = fmaf((v - mu) * rstd, sOG[d], sOB[d]);  // coalesced
    }
  }
}

extern "C" void kernel_launch(void* const* d_in, const int* in_sizes, int n_in,
                              void* d_out, int out_size, void* d_ws,
                              size_t ws_size, hipStream_t stream) {
  const float* x    = (const float*)d_in[0];
  const float* W    = (const float*)d_in[1];
  const float* b    = (const float*)d_in[2];
  const float* ln1g = (const float*)d_in[3];
  const float* ln1b = (const float*)d_in[4];
  const float* scw  = (const float*)d_in[5];
  const float* scb  = (const float*)d_in[6];
  const float* outg = (const float*)d_in[7];
  const float* outb = (const float*)d_in[8];

  float* out  = (float*)d_out;
  float* wout = out + (size_t)NTOK * DDIM;   // weights follow `out` flat
  float* st   = (float*)d_ws;                // 32 * 8 floats of LN stats

  vsn_stats<<<1, 32, 0, stream>>>(W, b, st);
  vsn_main<<<NTOK / BTOK, 256, 0, stream>>>(x, W, b, ln1g, ln1b, scw, scb,
                                            outg, outb, st, out, wout);
}